// MyLoss_10746008175287
// MI455X (gfx1250) — compile-verified
//
#include <hip/hip_runtime.h>

typedef float v2f __attribute__((ext_vector_type(2)));
typedef float v4f __attribute__((ext_vector_type(4)));
typedef float v8f __attribute__((ext_vector_type(8)));

#define MARGIN 0.1f

// Full 32-lane wave reduction using the CDNA5 matrix unit.
// A(16x4): lanes 0-15 carry M=0..15 for K=0,1; lanes 16-31 for K=2,3.
// With a = (acc, 0) and B = ones(4x16): D[m][n] = acc[m] + acc[m+16].
// C/D layout: VGPR r, lanes 0-15 = D[r][n]; lanes 16-31 = D[r+8][n].
// Sum the 8 result VGPRs, then one xor-16 shuffle-add finishes all 32 lanes.
__device__ __forceinline__ float wave_reduce_wmma(float acc) {
    v2f a; a.x = acc;  a.y = 0.0f;
    v2f b; b.x = 1.0f; b.y = 1.0f;
    v8f c = {0.f, 0.f, 0.f, 0.f, 0.f, 0.f, 0.f, 0.f};
    c = __builtin_amdgcn_wmma_f32_16x16x4_f32(
            /*neg_a=*/false, a, /*neg_b=*/false, b,
            /*c_mod=*/(short)0, c, /*reuse_a=*/false, /*reuse_b=*/false);
    float s = ((c[0] + c[1]) + (c[2] + c[3])) + ((c[4] + c[5]) + (c[6] + c[7]));
    s += __shfl_xor(s, 16, 32);
    return s;
}

// Block reduction; valid result only on threadIdx.x == 0.
__device__ __forceinline__ float block_reduce(float acc) {
    __shared__ float lds[32];
    float s = wave_reduce_wmma(acc);
    unsigned wid  = threadIdx.x >> 5;
    unsigned lane = threadIdx.x & 31u;
    if (lane == 0) lds[wid] = s;
    __syncthreads();
    float t = 0.0f;
    if (threadIdx.x == 0) {
        unsigned nw = (blockDim.x + 31u) >> 5;
        for (unsigned i = 0; i < nw; ++i) t += lds[i];
    }
    return t;
}

__device__ __forceinline__ float hinge4(v4f a, v4f b) {
    float h0 = fmaxf(0.0f, MARGIN - a.x + b.x);
    float h1 = fmaxf(0.0f, MARGIN - a.y + b.y);
    float h2 = fmaxf(0.0f, MARGIN - a.z + b.z);
    float h3 = fmaxf(0.0f, MARGIN - a.w + b.w);
    return (h0 + h1) + (h2 + h3);
}

__global__ void __launch_bounds__(256)
hinge_partial(const float* __restrict__ apm, const float* __restrict__ aorg,
              float* __restrict__ partial, int n) {
    const v4f* a4 = reinterpret_cast<const v4f*>(apm);
    const v4f* b4 = reinterpret_cast<const v4f*>(aorg);
    const int n4     = n >> 2;
    const int stride = gridDim.x * blockDim.x;

    float acc = 0.0f;
    int i = blockIdx.x * blockDim.x + threadIdx.x;

    // 4x-unrolled grid-stride: issue 8 independent NT b128 loads per wave
    // before the first s_wait_loadcnt — MLP is the limiter for a pure
    // HBM-streaming kernel (268MB / 23.3TB/s ~ 11.5us floor).
    for (; i + 3 * stride < n4; i += 4 * stride) {
        v4f a0 = __builtin_nontemporal_load(a4 + i);
        v4f a1 = __builtin_nontemporal_load(a4 + i + stride);
        v4f a2 = __builtin_nontemporal_load(a4 + i + 2 * stride);
        v4f a3 = __builtin_nontemporal_load(a4 + i + 3 * stride);
        v4f b0 = __builtin_nontemporal_load(b4 + i);
        v4f b1 = __builtin_nontemporal_load(b4 + i + stride);
        v4f b2 = __builtin_nontemporal_load(b4 + i + 2 * stride);
        v4f b3 = __builtin_nontemporal_load(b4 + i + 3 * stride);
        acc += hinge4(a0, b0);
        acc += hinge4(a1, b1);
        acc += hinge4(a2, b2);
        acc += hinge4(a3, b3);
    }
    // Remainder float4 iterations.
    for (; i < n4; i += stride) {
        v4f a = __builtin_nontemporal_load(a4 + i);
        v4f b = __builtin_nontemporal_load(b4 + i);
        acc += hinge4(a, b);
    }
    // Scalar tail (empty when n % 4 == 0; kept for generality).
    for (int j = (n4 << 2) + blockIdx.x * blockDim.x + threadIdx.x; j < n; j += stride)
        acc += fmaxf(0.0f, MARGIN - apm[j] + aorg[j]);

    float bsum = block_reduce(acc);
    if (threadIdx.x == 0) partial[blockIdx.x] = bsum;
}

__global__ void __launch_bounds__(256)
hinge_finalize(const float* __restrict__ partial, int nparts, float inv_n,
               float* __restrict__ out) {
    float acc = 0.0f;
    for (int i = threadIdx.x; i < nparts; i += blockDim.x)
        acc += partial[i];
    float s = block_reduce(acc);
    if (threadIdx.x == 0) out[0] = s * inv_n;
}

extern "C" void kernel_launch(void* const* d_in, const int* in_sizes, int n_in,
                              void* d_out, int out_size, void* d_ws, size_t ws_size,
                              hipStream_t stream) {
    const float* apm  = (const float*)d_in[0];  // alphas_part_max
    const float* aorg = (const float*)d_in[1];  // alphas_org
    float* out        = (float*)d_out;
    float* partial    = (float*)d_ws;           // 2048 floats = 8 KB scratch
    const int n = in_sizes[0];

    const int THREADS = 256;   // 8 wave32 waves per block
    const int BLOCKS  = 2048;  // exactly tiles N=2^25 at 4 unrolled float4/thread

    hinge_partial<<<BLOCKS, THREADS, 0, stream>>>(apm, aorg, partial, n);
    hinge_finalize<<<1, THREADS, 0, stream>>>(partial, BLOCKS, 1.0f / (float)n, out);
}